// TripletRankingLoss_8057358647420
// MI455X (gfx1250) — compile-verified
//
#include <hip/hip_runtime.h>
#include <hip/hip_bf16.h>

typedef __attribute__((ext_vector_type(2))) float v2f;
typedef __attribute__((ext_vector_type(8))) float v8f;

#define N_ROWS  16384
#define EDIM    1024
#define KSLICES 4
#define SLICE_K (EDIM / KSLICES)   /* 256 */
#define GROUPS  (N_ROWS / 16)      /* 1024 */
#define MARGIN  1.0f

// Select element s of a v8f accumulator (unrolled cndmask chain; s is per-lane).
__device__ __forceinline__ float pick8(v8f a, int s) {
  float r = a[0];
  r = (s == 1) ? a[1] : r;
  r = (s == 2) ? a[2] : r;
  r = (s == 3) ? a[3] : r;
  r = (s == 4) ? a[4] : r;
  r = (s == 5) ? a[5] : r;
  r = (s == 6) ? a[6] : r;
  r = (s == 7) ? a[7] : r;
  return r;
}

// Kernel 1: each wave owns (16-row group, K-slice). Three f32 WMMA accumulators
// hold the 16x16 products whose diagonals are the paired dot products.
// A(16x4 f32) and B(4x16 f32) operands share the same per-lane layout:
//   lane l holds row (l & 15), K-columns kb + 2*(l>>4) .. +1  -> one float2 load.
__global__ void __launch_bounds__(256)
triplet_dots_wmma(const float* __restrict__ audio,
                  const float* __restrict__ query,
                  const int*   __restrict__ aneg,
                  const int*   __restrict__ qneg,
                  float* __restrict__ anchorP,   // [KSLICES][N_ROWS]
                  float* __restrict__ aimpP,     // [KSLICES][N_ROWS]
                  float* __restrict__ qimpP)     // [KSLICES][N_ROWS]
{
  const int lane  = threadIdx.x & 31;
  const int wid   = blockIdx.x * (blockDim.x >> 5) + (threadIdx.x >> 5);
  const int group = wid >> 2;            // 0..GROUPS-1
  const int slice = wid & (KSLICES - 1); // 0..3
  const int base  = group << 4;
  const int row   = lane & 15;
  const int koff  = (lane >> 4) << 1;    // 0 for lanes 0-15, 2 for lanes 16-31
  const int r     = base + row;

  const float* pa  = audio + (size_t)r        * EDIM + koff;
  const float* pq  = query + (size_t)r        * EDIM + koff;
  const float* pan = audio + (size_t)aneg[r]  * EDIM + koff;
  const float* pqn = query + (size_t)qneg[r]  * EDIM + koff;

  v8f acc_anchor = {};
  v8f acc_aimp   = {};
  v8f acc_qimp   = {};

  int kb = slice * SLICE_K;
  #pragma unroll 8
  for (int it = 0; it < SLICE_K / 4; ++it, kb += 4) {
    v2f va  = *(const v2f*)(pa  + kb);   // A rows of audio
    v2f vq  = *(const v2f*)(pq  + kb);   // B = (query rows)^T
    v2f van = *(const v2f*)(pan + kb);   // A rows of gathered audio
    v2f vqn = *(const v2f*)(pqn + kb);   // B = (gathered query rows)^T
    // D += A x B ; diag(D)[m] accumulates <row_m, row_m> pairs.
    acc_anchor = __builtin_amdgcn_wmma_f32_16x16x4_f32(
        false, va,  false, vq,  (short)0, acc_anchor, false, false);
    acc_aimp   = __builtin_amdgcn_wmma_f32_16x16x4_f32(
        false, van, false, vq,  (short)0, acc_aimp,   false, false);
    acc_qimp   = __builtin_amdgcn_wmma_f32_16x16x4_f32(
        false, va,  false, vqn, (short)0, acc_qimp,   false, false);
  }

  // Diagonal lives at: rows 0..7 -> lane m, VGPR m ; rows 8..15 -> lane m+16, VGPR m-8.
  const bool act  = (lane < 8) || (lane >= 24);
  const int  sel  = (lane < 8) ? lane : ((lane >= 24) ? (lane - 24) : 0);
  const int  drow = (lane < 8) ? lane : (lane - 16);

  if (act) {
    const size_t idx = (size_t)slice * N_ROWS + (size_t)(base + drow);
    anchorP[idx] = pick8(acc_anchor, sel);
    aimpP[idx]   = pick8(acc_aimp,   sel);
    qimpP[idx]   = pick8(acc_qimp,   sel);
  }
}

// Kernel 2: per-row hinge loss from the 4 K-slice partials, block tree-reduce.
__global__ void __launch_bounds__(256)
triplet_loss_reduce(const float* __restrict__ anchorP,
                    const float* __restrict__ aimpP,
                    const float* __restrict__ qimpP,
                    float* __restrict__ partials)   // [gridDim.x]
{
  __shared__ float red[256];
  const int i = blockIdx.x * 256 + threadIdx.x;

  float anc = anchorP[i] + anchorP[N_ROWS + i] + anchorP[2 * N_ROWS + i] + anchorP[3 * N_ROWS + i];
  float aim = aimpP[i]   + aimpP[N_ROWS + i]   + aimpP[2 * N_ROWS + i]   + aimpP[3 * N_ROWS + i];
  float qim = qimpP[i]   + qimpP[N_ROWS + i]   + qimpP[2 * N_ROWS + i]   + qimpP[3 * N_ROWS + i];

  float loss = fmaxf(MARGIN + qim - anc, 0.0f) + fmaxf(MARGIN + aim - anc, 0.0f);

  red[threadIdx.x] = loss;
  __syncthreads();
  #pragma unroll
  for (int s = 128; s > 0; s >>= 1) {
    if (threadIdx.x < s) red[threadIdx.x] += red[threadIdx.x + s];
    __syncthreads();
  }
  if (threadIdx.x == 0) partials[blockIdx.x] = red[0];
}

// Kernel 3: sum the 64 block partials -> mean. Deterministic, no atomics.
__global__ void __launch_bounds__(64)
triplet_final(const float* __restrict__ partials, float* __restrict__ out)
{
  __shared__ float red[64];
  red[threadIdx.x] = partials[threadIdx.x];
  __syncthreads();
  #pragma unroll
  for (int s = 32; s > 0; s >>= 1) {
    if (threadIdx.x < s) red[threadIdx.x] += red[threadIdx.x + s];
    __syncthreads();
  }
  if (threadIdx.x == 0) out[0] = red[0] / (float)N_ROWS;
}

extern "C" void kernel_launch(void* const* d_in, const int* in_sizes, int n_in,
                              void* d_out, int out_size, void* d_ws, size_t ws_size,
                              hipStream_t stream) {
  const float* audio = (const float*)d_in[0];
  const float* query = (const float*)d_in[1];
  const int*   aneg  = (const int*)d_in[2];
  const int*   qneg  = (const int*)d_in[3];
  float*       out   = (float*)d_out;

  // Workspace layout (floats): 3 x [KSLICES][N_ROWS] dot partials + 64 block partials.
  float* ws       = (float*)d_ws;
  float* anchorP  = ws;                           // 4*16384
  float* aimpP    = ws + KSLICES * N_ROWS;        // 4*16384
  float* qimpP    = ws + 2 * KSLICES * N_ROWS;    // 4*16384
  float* partials = ws + 3 * KSLICES * N_ROWS;    // 64
  (void)ws_size; (void)in_sizes; (void)n_in; (void)out_size;

  // 4096 waves (1024 groups x 4 K-slices), 8 waves per 256-thread block.
  const int waves  = GROUPS * KSLICES;
  const int blocks = waves / 8;
  triplet_dots_wmma<<<blocks, 256, 0, stream>>>(audio, query, aneg, qneg,
                                                anchorP, aimpP, qimpP);
  triplet_loss_reduce<<<N_ROWS / 256, 256, 0, stream>>>(anchorP, aimpP, qimpP, partials);
  triplet_final<<<1, 64, 0, stream>>>(partials, out);
}